// FeaturelessSparseCloudConvolution_67173288509588
// MI455X (gfx1250) — compile-verified
//
#include <hip/hip_runtime.h>
#include <hip/hip_bf16.h>

typedef __attribute__((ext_vector_type(2))) float v2f;
typedef __attribute__((ext_vector_type(8))) float v8f;

#define T_FEAT 8

// ---------------------------------------------------------------------------
// Kernel A: zero the (N x 8) f32 accumulator in workspace (8 MB). Must run
// every call since the harness does not re-poison d_ws between replays.
// ---------------------------------------------------------------------------
__global__ void fscc_zero_kernel(float4* __restrict__ p, int n4) {
    int i = blockIdx.x * blockDim.x + threadIdx.x;
    if (i < n4) p[i] = make_float4(0.f, 0.f, 0.f, 0.f);
}

// ---------------------------------------------------------------------------
// Kernel B: scatter-add (segment_sum). One thread per edge.
//   edge_features: (T, E) row-major  -> ef[t*E + e]  (coalesced across e)
//   sparse_indices: (E, 2) int64, row = idx[e*2 + 0]
//   agg: (N, T) row-major f32 accumulator, updated with f32 global atomics.
// The 8 MB accumulator stays resident in the 192 MB L2, so the 64M atomic
// fadds resolve at L2 rate; HBM traffic is dominated by the 384 MB of inputs.
// ---------------------------------------------------------------------------
__global__ void fscc_scatter_kernel(const float* __restrict__ ef,
                                    const long long* __restrict__ idx,
                                    float* __restrict__ agg,
                                    int E) {
    int e = blockIdx.x * blockDim.x + threadIdx.x;
    if (e >= E) return;
    int row = (int)idx[(size_t)e * 2];  // column 0 of (E,2)
    float* dst = agg + (size_t)row * T_FEAT;
#pragma unroll
    for (int t = 0; t < T_FEAT; ++t) {
        float v = ef[(size_t)t * (size_t)E + (size_t)e];
        unsafeAtomicAdd(dst + t, v);   // -> global_atomic_add_f32 (no-return)
    }
}

// ---------------------------------------------------------------------------
// Kernel C: out = agg(N x 8) @ W(8 x 64) + bias, via V_WMMA_F32_16X16X4_F32.
// One wave per 16-row strip; 4 column tiles of 16; K=8 as two K=4 WMMAs.
//
// f32 A-matrix 16x4 layout (wave32): lane m = lane&15, kb = (lane>>4)*2;
//   a[0] = A[m][kb], a[1] = A[m][kb+1]  -> contiguous float2 from agg.
// f32 B-matrix 4x16: mirrored striping with n = lane&15:
//   b[0] = B[kb][n], b[1] = B[kb+1][n].
// f32 C/D 16x16: VGPR v holds row v (lanes 0-15) and row v+8 (lanes 16-31),
//   col = lane&15 within the tile.
// ---------------------------------------------------------------------------
__global__ void fscc_gemm_kernel(const float* __restrict__ agg,
                                 const float* __restrict__ W,
                                 const float* __restrict__ bias,
                                 float* __restrict__ out,
                                 int N) {
    const int lane = threadIdx.x & 31;
    const int wave = blockIdx.x * (blockDim.x >> 5) + (threadIdx.x >> 5);
    const int r0 = wave * 16;
    if (r0 >= N) return;

    const int m  = lane & 15;
    const int hi = lane >> 4;        // 0 for lanes 0-15, 1 for lanes 16-31
    const int kb = hi * 2;           // K base within a K=4 chunk: 0 or 2

    // A tiles for the two K-chunks (k in [0,4) and [4,8))
    const float* arow = agg + (size_t)(r0 + m) * T_FEAT;
    v2f a0 = *(const v2f*)(arow + kb);        // K = kb, kb+1
    v2f a1 = *(const v2f*)(arow + 4 + kb);    // K = 4+kb, 4+kb+1

#pragma unroll
    for (int ct = 0; ct < 4; ++ct) {
        const int n  = ct * 16 + m;           // global output column
        v2f b0, b1;
        b0.x = W[(kb + 0) * 64 + n];
        b0.y = W[(kb + 1) * 64 + n];
        b1.x = W[(4 + kb + 0) * 64 + n];
        b1.y = W[(4 + kb + 1) * 64 + n];

        v8f c = {};
        c = __builtin_amdgcn_wmma_f32_16x16x4_f32(false, a0, false, b0,
                                                  (short)0, c, false, false);
        c = __builtin_amdgcn_wmma_f32_16x16x4_f32(false, a1, false, b1,
                                                  (short)0, c, false, false);

        const float bv = bias[n];
#pragma unroll
        for (int v = 0; v < 8; ++v) {
            const int row = r0 + v + hi * 8;
            out[(size_t)row * 64 + n] = c[v] + bv;
        }
    }
}

// ---------------------------------------------------------------------------
// Launch: zero(agg) -> scatter -> wmma gemm, all stream-ordered.
// Inputs: [0] edge_features (T*E f32), [1] sparse_indices (E*2 i64),
//         [2] out_size scalar, [3] kernel (T*F f32), [4] bias (F f32)
// ---------------------------------------------------------------------------
extern "C" void kernel_launch(void* const* d_in, const int* in_sizes, int n_in,
                              void* d_out, int out_size, void* d_ws, size_t ws_size,
                              hipStream_t stream) {
    const float*     ef   = (const float*)d_in[0];
    const long long* idx  = (const long long*)d_in[1];
    const float*     W    = (const float*)d_in[3];
    const float*     bias = (const float*)d_in[4];
    float*           out  = (float*)d_out;
    float*           agg  = (float*)d_ws;      // N*8 floats = 8 MB scratch

    const int E = in_sizes[0] / T_FEAT;        // 8388608
    const int N = out_size / 64;               // 262144

    // A: zero accumulator (N*8 floats, vectorized as float4)
    {
        int n4 = (N * T_FEAT) / 4;
        int blocks = (n4 + 255) / 256;
        fscc_zero_kernel<<<blocks, 256, 0, stream>>>((float4*)agg, n4);
    }
    // B: scatter-add edges into agg
    {
        int blocks = (E + 255) / 256;
        fscc_scatter_kernel<<<blocks, 256, 0, stream>>>(ef, idx, agg, E);
    }
    // C: WMMA GEMM (one wave per 16 rows; 8 waves per 256-thread block)
    {
        int waves  = N / 16;                   // 16384
        int blocks = (waves + 7) / 8;          // 2048
        fscc_gemm_kernel<<<blocks, 256, 0, stream>>>(agg, W, bias, out, N);
    }
}